// Memory_59777354826390
// MI455X (gfx1250) — compile-verified
//
#include <hip/hip_runtime.h>
#include <math.h>

// DNC memory forward step for MI455X (gfx1250, wave32).
// B=64, N=1024, W=64, R=4, NW=1. All f32.
// Dominant cost: streaming prev_link (256 MB) once -> ~12us HBM floor.
// WMMA f32 16x16x4 used for read_words = read_weights @ memory_new.

constexpr int kB = 64;
constexpr int kN = 1024;
constexpr int kW = 64;
constexpr int kR = 4;
constexpr float kEPS = 1e-6f;

typedef __attribute__((ext_vector_type(2))) float v2f;
typedef __attribute__((ext_vector_type(8))) float v8f;

__device__ __forceinline__ float softplusf(float x) {
  return (x > 20.f) ? x : log1pf(expf(x));
}

__device__ __forceinline__ float block_reduce_max(float v, float* s, int tid) {
  s[tid] = v; __syncthreads();
  for (int off = 512; off > 0; off >>= 1) {
    if (tid < off) s[tid] = fmaxf(s[tid], s[tid + off]);
    __syncthreads();
  }
  float r = s[0]; __syncthreads();
  return r;
}

__device__ __forceinline__ float block_reduce_sum(float v, float* s, int tid) {
  s[tid] = v; __syncthreads();
  for (int off = 512; off > 0; off >>= 1) {
    if (tid < off) s[tid] += s[tid + off];
    __syncthreads();
  }
  float r = s[0]; __syncthreads();
  return r;
}

// ---------------------------------------------------------------------------
// Stage 1: per-batch block (1024 threads, one per memory slot n).
// usage update, write content softmax, allocation (bitonic sort + cumprod),
// write weights, memory_new, read content softmax.
// ---------------------------------------------------------------------------
__global__ __launch_bounds__(1024)
void dnc_stage1(const float* __restrict__ memory,        // B,N,W
                const float* __restrict__ prev_read_w,   // B,R,N
                const float* __restrict__ prev_write_w,  // B,1,N
                const float* __restrict__ prev_usage,    // B,N
                const float* __restrict__ read_keys,     // B,R,W
                const float* __restrict__ read_str,      // B,R
                const float* __restrict__ write_keys,    // B,1,W
                const float* __restrict__ write_str,     // B,1
                const float* __restrict__ write_vec,     // B,1,W
                const float* __restrict__ erase_vec,     // B,1,W
                const float* __restrict__ free_gate,     // B,R
                const float* __restrict__ alloc_gate,    // B,1
                const float* __restrict__ write_gate,    // B,1
                float* __restrict__ ws_ww,               // B,N
                float* __restrict__ ws_memnew,           // B,N,W
                float* __restrict__ ws_rc)               // B,R,N
{
  __shared__ float s_a[kN];
  __shared__ float s_b[kN];
  __shared__ float s_c[kN];
  __shared__ int   s_i[kN];
  __shared__ float s_wk[kW], s_wv[kW], s_ev[kW];
  __shared__ float s_rk[kR][kW];
  __shared__ float s_krn[kR], s_sp[kR];
  __shared__ float s_wsp;

  const int n = threadIdx.x;
  const int b = blockIdx.x;

  if (n < kW) {
    s_wk[n] = write_keys[b * kW + n];
    s_wv[n] = write_vec[b * kW + n];
    s_ev[n] = erase_vec[b * kW + n];
  }
  if (n < kR * kW) s_rk[n / kW][n % kW] = read_keys[(size_t)b * kR * kW + n];
  if (n < kR) {
    float sum2 = 0.f;
    for (int k = 0; k < kW; ++k) {
      float v = read_keys[((size_t)b * kR + n) * kW + k];
      sum2 += v * v;
    }
    s_krn[n] = sqrtf(sum2 + kEPS);
    s_sp[n] = softplusf(read_str[b * kR + n]);
  }
  if (n == 0) s_wsp = softplusf(write_str[b]);
  __syncthreads();

  // ---- write content weights (cosine + softmax over N) on OLD memory ----
  const float* mrow = memory + ((size_t)b * kN + n) * kW;
  float dot = 0.f, mn2 = 0.f, kn2 = 0.f;
  for (int k = 0; k < kW; ++k) {
    float m = mrow[k], w = s_wk[k];
    dot += m * w; mn2 += m * m; kn2 += w * w;
  }
  float sim = dot / (sqrtf(kn2 + kEPS) * sqrtf(mn2 + kEPS) + kEPS);
  float sharp = sim * s_wsp;
  float mx = block_reduce_max(sharp, s_b, n);
  float e = expf(sharp - mx);
  float esum = block_reduce_sum(e, s_b, n);
  float wc = e / esum;

  // ---- usage update ----
  float pu = prev_usage[b * kN + n];
  float aw = prev_write_w[b * kN + n];            // NW=1: agg_write == ww
  float usage = pu + (1.f - pu) * aw;
  float phi = 1.f;
  #pragma unroll
  for (int r = 0; r < kR; ++r)
    phi *= (1.f - free_gate[b * kR + r] * prev_read_w[((size_t)b * kR + r) * kN + n]);
  usage *= phi;

  // ---- allocation: sort nonusage desc (stable), exclusive cumprod, scatter ----
  float uu = kEPS + (1.f - kEPS) * usage;
  s_a[n] = 1.f - uu;
  s_i[n] = n;
  __syncthreads();
  for (int k = 2; k <= kN; k <<= 1) {
    for (int j = k >> 1; j > 0; j >>= 1) {
      int ixj = n ^ j;
      if (ixj > n) {
        float va = s_a[n], vb = s_a[ixj];
        int ia = s_i[n], ib = s_i[ixj];
        bool afirst = (va > vb) || (va == vb && ia < ib);  // descending, stable
        bool dir = ((n & k) == 0);
        if (afirst != dir) {
          s_a[n] = vb; s_a[ixj] = va;
          s_i[n] = ib; s_i[ixj] = ia;
        }
      }
      __syncthreads();
    }
  }
  // inclusive product scan of sorted_usage (Hillis-Steele, ping-pong)
  float su = 1.f - s_a[n];
  float* ps = s_b; float* pd = s_c;
  ps[n] = su; __syncthreads();
  for (int off = 1; off < kN; off <<= 1) {
    float v = ps[n];
    if (n >= off) v *= ps[n - off];
    pd[n] = v; __syncthreads();
    float* tswap = ps; ps = pd; pd = tswap;
  }
  float pe = (n == 0) ? 1.f : ps[n - 1];   // exclusive cumprod
  float salloc = s_a[n] * pe;              // sorted_nonusage * prod_excl
  pd[s_i[n]] = salloc;                     // scatter back to original order
  __syncthreads();
  float alloc = pd[n];

  // ---- write weights (NW=1) ----
  float ag = alloc_gate[b];
  float wg = write_gate[b];
  float wwn = wg * (ag * alloc + (1.f - ag) * wc);
  ws_ww[b * kN + n] = wwn;

  // ---- memory_new + read-content dot products ----
  float mn2n = 0.f;
  float rd0 = 0.f, rd1 = 0.f, rd2 = 0.f, rd3 = 0.f;
  float* mnrow = ws_memnew + ((size_t)b * kN + n) * kW;
  for (int k = 0; k < kW; ++k) {
    float m = mrow[k];
    float nm = m * (1.f - wwn * s_ev[k]) + wwn * s_wv[k];
    mnrow[k] = nm;
    mn2n += nm * nm;
    rd0 += s_rk[0][k] * nm; rd1 += s_rk[1][k] * nm;
    rd2 += s_rk[2][k] * nm; rd3 += s_rk[3][k] * nm;
  }
  float imn = sqrtf(mn2n + kEPS);
  float rds[kR] = {rd0, rd1, rd2, rd3};
  __syncthreads();  // done with pd before reusing reduction scratch

  for (int r = 0; r < kR; ++r) {
    float sim_r = rds[r] / (s_krn[r] * imn + kEPS);
    float sh = sim_r * s_sp[r];
    float mx_r = block_reduce_max(sh, s_b, n);
    float e_r = expf(sh - mx_r);
    float sum_r = block_reduce_sum(e_r, s_b, n);
    ws_rc[((size_t)b * kR + r) * kN + n] = e_r / sum_r;
  }
}

// ---------------------------------------------------------------------------
__global__ void dnc_zero(float* __restrict__ p, int count) {
  int i = blockIdx.x * blockDim.x + threadIdx.x;
  if (i < count) p[i] = 0.f;
}

// ---------------------------------------------------------------------------
// Fused link construction + fwd/bwd contractions; streams prev_link once.
//   l(n,m) = (n!=m) * ((1 - ww[n] - ww[m])*PL[n,m] + ww[n]*prec[m])
//   fwd[r,n] += prw[r,m]*l     bwd[r,m] += prw[r,n]*l
// Grid: (colTiles=2, rowTiles=8, B). Block: 256 threads = 8 waves x 32 lanes.
// Each wave owns rows (ty + 8*jr); lanes cover 128 consecutive floats as
// float4 -> fully coalesced 512B reads. fwd reduced in-wave via shfl;
// bwd accumulated in registers per-lane then merged via LDS atomics.
// ---------------------------------------------------------------------------
__global__ __launch_bounds__(256)
void dnc_link(const float* __restrict__ prev_link,   // B,1,N,N
              const float* __restrict__ prev_prec,   // B,1,N
              const float* __restrict__ prev_read_w, // B,R,N
              const float* __restrict__ ws_ww,       // B,N
              float* __restrict__ ws_fwd,            // B,R,N
              float* __restrict__ ws_bwd)            // B,R,N
{
  constexpr int RT = 128, CT = 512;
  __shared__ float s_wwr[RT];
  __shared__ float s_prwr[kR][RT];
  __shared__ float s_wwc[CT];
  __shared__ float s_prc[CT];
  __shared__ float s_prwc[kR][CT];
  __shared__ float s_bacc[kR][CT];

  const int b = blockIdx.z;
  const int n0 = blockIdx.y * RT;
  const int m0 = blockIdx.x * CT;
  const int t = threadIdx.x;

  for (int i = t; i < RT; i += 256) {
    s_wwr[i] = ws_ww[b * kN + n0 + i];
    #pragma unroll
    for (int r = 0; r < kR; ++r)
      s_prwr[r][i] = prev_read_w[((size_t)b * kR + r) * kN + n0 + i];
  }
  for (int i = t; i < CT; i += 256) {
    s_wwc[i] = ws_ww[b * kN + m0 + i];
    s_prc[i] = prev_prec[b * kN + m0 + i];
    #pragma unroll
    for (int r = 0; r < kR; ++r) {
      s_prwc[r][i] = prev_read_w[((size_t)b * kR + r) * kN + m0 + i];
      s_bacc[r][i] = 0.f;
    }
  }
  __syncthreads();

  const int tx = t & 31, ty = t >> 5;
  float bl0[16], bl1[16], bl2[16], bl3[16];
  #pragma unroll
  for (int i = 0; i < 16; ++i) { bl0[i] = 0.f; bl1[i] = 0.f; bl2[i] = 0.f; bl3[i] = 0.f; }

  for (int jr = 0; jr < 16; ++jr) {
    const int nl = ty + 8 * jr;
    const int n = n0 + nl;
    const float wwn = s_wwr[nl];
    const float pr0 = s_prwr[0][nl], pr1 = s_prwr[1][nl];
    const float pr2 = s_prwr[2][nl], pr3 = s_prwr[3][nl];
    float f0 = 0.f, f1 = 0.f, f2 = 0.f, f3 = 0.f;
    const float* plrow = prev_link + ((size_t)b * kN + n) * kN + m0;
    // Prefetch this wave's next row (8 rows ahead) into cache (global_prefetch_b8).
    if (jr < 15) __builtin_prefetch(plrow + (size_t)8 * kN + (size_t)tx * 16, 0, 1);
    #pragma unroll
    for (int i = 0; i < 4; ++i) {
      const int mb = (tx + 32 * i) * 4;
      const float4 pl = *(const float4*)(plrow + mb);
      const float plv[4] = {pl.x, pl.y, pl.z, pl.w};
      #pragma unroll
      for (int c = 0; c < 4; ++c) {
        const int ml = mb + c;
        const int m = m0 + ml;
        float l = (n == m) ? 0.f
                           : ((1.f - wwn - s_wwc[ml]) * plv[c] + wwn * s_prc[ml]);
        f0 += s_prwc[0][ml] * l; f1 += s_prwc[1][ml] * l;
        f2 += s_prwc[2][ml] * l; f3 += s_prwc[3][ml] * l;
        const int ci = i * 4 + c;
        bl0[ci] += pr0 * l; bl1[ci] += pr1 * l;
        bl2[ci] += pr2 * l; bl3[ci] += pr3 * l;
      }
    }
    #pragma unroll
    for (int off = 16; off > 0; off >>= 1) {
      f0 += __shfl_down(f0, off, 32);
      f1 += __shfl_down(f1, off, 32);
      f2 += __shfl_down(f2, off, 32);
      f3 += __shfl_down(f3, off, 32);
    }
    if (tx == 0) {
      atomicAdd(&ws_fwd[((size_t)b * kR + 0) * kN + n], f0);
      atomicAdd(&ws_fwd[((size_t)b * kR + 1) * kN + n], f1);
      atomicAdd(&ws_fwd[((size_t)b * kR + 2) * kN + n], f2);
      atomicAdd(&ws_fwd[((size_t)b * kR + 3) * kN + n], f3);
    }
  }
  #pragma unroll
  for (int i = 0; i < 4; ++i) {
    #pragma unroll
    for (int c = 0; c < 4; ++c) {
      const int ml = (tx + 32 * i) * 4 + c;
      const int ci = i * 4 + c;
      atomicAdd(&s_bacc[0][ml], bl0[ci]);
      atomicAdd(&s_bacc[1][ml], bl1[ci]);
      atomicAdd(&s_bacc[2][ml], bl2[ci]);
      atomicAdd(&s_bacc[3][ml], bl3[ci]);
    }
  }
  __syncthreads();
  for (int i = t; i < CT; i += 256)
    for (int r = 0; r < kR; ++r)
      atomicAdd(&ws_bwd[((size_t)b * kR + r) * kN + m0 + i], s_bacc[r][i]);
}

// ---------------------------------------------------------------------------
// Final read: mode softmax, read_weights, then
//   read_words(16pad x 64) = read_weights(16 x 1024) @ memory_new(1024 x 64)
// via V_WMMA_F32_16X16X4_F32, one wave per batch, K-loop of 256 steps,
// four 16-wide N-tiles over W=64.
// A layout (32b 16x4): lanes 0-15 M=lane, VGPR0=K0, VGPR1=K1; lanes 16-31 K2/K3.
// B layout (32b 4x16): VGPR0 lanes0-15 row K0, lanes16-31 row K2; VGPR1 K1/K3.
// D layout (32b 16x16): VGPR v -> M=v (lanes 0-15), M=v+8 (lanes 16-31).
// s_rw has a 5th all-zero row so lanes with M>=4 read zeros unconditionally:
// keeps EXEC all-ones with no saveexec-guarded ds_loads in the hot loop, and
// lets the A operand be fetched as a single ds_load_b64.
// ---------------------------------------------------------------------------
__global__ __launch_bounds__(32)
void dnc_read(const float* __restrict__ read_mode,  // B,R,3
              const float* __restrict__ ws_rc,      // B,R,N
              const float* __restrict__ ws_fwd,     // B,R,N
              const float* __restrict__ ws_bwd,     // B,R,N
              const float* __restrict__ ws_memnew,  // B,N,W
              float* __restrict__ out)              // B,R,W
{
  __shared__ float s_rw[kR + 1][kN];   // row kR is all zeros (padding rows)
  __shared__ float s_mode[kR][3];
  const int b = blockIdx.x;
  const int lane = threadIdx.x;

  if (lane < kR) {
    float m0 = read_mode[((size_t)b * kR + lane) * 3 + 0];
    float m1 = read_mode[((size_t)b * kR + lane) * 3 + 1];
    float m2 = read_mode[((size_t)b * kR + lane) * 3 + 2];
    float mx = fmaxf(m0, fmaxf(m1, m2));
    float e0 = expf(m0 - mx), e1 = expf(m1 - mx), e2 = expf(m2 - mx);
    float s = e0 + e1 + e2;
    s_mode[lane][0] = e0 / s;  // backward
    s_mode[lane][1] = e1 / s;  // forward
    s_mode[lane][2] = e2 / s;  // content
  }
  __syncthreads();

  for (int i = lane; i < kN; i += 32) {
    #pragma unroll
    for (int r = 0; r < kR; ++r) {
      size_t idx = ((size_t)b * kR + r) * kN + i;
      s_rw[r][i] = s_mode[r][2] * ws_rc[idx]
                 + s_mode[r][1] * ws_fwd[idx]
                 + s_mode[r][0] * ws_bwd[idx];
    }
    s_rw[kR][i] = 0.f;   // zero row for padded A rows M=4..15
  }
  __syncthreads();

  v8f acc0 = {}; v8f acc1 = {}; v8f acc2 = {}; v8f acc3 = {};
  const int rA = lane & 15;
  const int rowsel = (rA < kR) ? rA : kR;      // rows >=4 read the zero row
  const int koff = (lane < 16) ? 0 : 2;
  const int col = lane & 15;
  const float* arow = &s_rw[rowsel][0];
  const float* mn = ws_memnew + (size_t)b * kN * kW;

  for (int k0 = 0; k0 < kN; k0 += 4) {
    v2f a = *(const v2f*)(arow + k0 + koff);   // single ds_load_b64, EXEC all-1s
    const float* r0 = mn + (size_t)(k0 + koff) * kW;
    const float* r1 = r0 + kW;
    v2f b0, b1, b2, b3;
    b0.x = r0[col + 0];  b0.y = r1[col + 0];
    b1.x = r0[col + 16]; b1.y = r1[col + 16];
    b2.x = r0[col + 32]; b2.y = r1[col + 32];
    b3.x = r0[col + 48]; b3.y = r1[col + 48];
    acc0 = __builtin_amdgcn_wmma_f32_16x16x4_f32(false, a, false, b0, (short)0, acc0, false, false);
    acc1 = __builtin_amdgcn_wmma_f32_16x16x4_f32(false, a, false, b1, (short)0, acc1, false, false);
    acc2 = __builtin_amdgcn_wmma_f32_16x16x4_f32(false, a, false, b2, (short)0, acc2, false, false);
    acc3 = __builtin_amdgcn_wmma_f32_16x16x4_f32(false, a, false, b3, (short)0, acc3, false, false);
  }

  if (lane < 16) {
    #pragma unroll
    for (int v = 0; v < kR; ++v) {   // only M=0..3 are real read heads
      out[((size_t)b * kR + v) * kW + 0 + lane]  = acc0[v];
      out[((size_t)b * kR + v) * kW + 16 + lane] = acc1[v];
      out[((size_t)b * kR + v) * kW + 32 + lane] = acc2[v];
      out[((size_t)b * kR + v) * kW + 48 + lane] = acc3[v];
    }
  }
}

// ---------------------------------------------------------------------------
extern "C" void kernel_launch(void* const* d_in, const int* in_sizes, int n_in,
                              void* d_out, int out_size, void* d_ws, size_t ws_size,
                              hipStream_t stream) {
  (void)in_sizes; (void)n_in; (void)out_size; (void)ws_size;

  const float* memory       = (const float*)d_in[0];
  const float* prev_read_w  = (const float*)d_in[1];
  const float* prev_write_w = (const float*)d_in[2];
  const float* prev_usage   = (const float*)d_in[3];
  const float* prev_link    = (const float*)d_in[4];
  const float* prev_prec    = (const float*)d_in[5];
  const float* read_keys    = (const float*)d_in[6];
  const float* read_str     = (const float*)d_in[7];
  const float* write_keys   = (const float*)d_in[8];
  const float* write_str    = (const float*)d_in[9];
  const float* write_vec    = (const float*)d_in[10];
  const float* erase_vec    = (const float*)d_in[11];
  const float* free_gate    = (const float*)d_in[12];
  const float* alloc_gate   = (const float*)d_in[13];
  const float* write_gate   = (const float*)d_in[14];
  const float* read_mode    = (const float*)d_in[15];

  // Workspace layout (floats): ww | memnew | rc | fwd | bwd  (~20.2 MB total)
  float* ws        = (float*)d_ws;
  float* ws_ww     = ws;
  float* ws_memnew = ws_ww + (size_t)kB * kN;
  float* ws_rc     = ws_memnew + (size_t)kB * kN * kW;
  float* ws_fwd    = ws_rc + (size_t)kB * kR * kN;
  float* ws_bwd    = ws_fwd + (size_t)kB * kR * kN;

  dnc_stage1<<<kB, kN, 0, stream>>>(memory, prev_read_w, prev_write_w, prev_usage,
                                    read_keys, read_str, write_keys, write_str,
                                    write_vec, erase_vec, free_gate, alloc_gate,
                                    write_gate, ws_ww, ws_memnew, ws_rc);

  const int zcount = 2 * kB * kR * kN;  // fwd + bwd contiguous
  dnc_zero<<<(zcount + 255) / 256, 256, 0, stream>>>(ws_fwd, zcount);

  dnc_link<<<dim3(2, 8, kB), 256, 0, stream>>>(prev_link, prev_prec, prev_read_w,
                                               ws_ww, ws_fwd, ws_bwd);

  dnc_read<<<kB, 32, 0, stream>>>(read_mode, ws_rc, ws_fwd, ws_bwd, ws_memnew,
                                  (float*)d_out);
}